// FacialGAT_62208306315392
// MI455X (gfx1250) — compile-verified
//
#include <hip/hip_runtime.h>
#include <math.h>

#define NN   100000
#define EE   1600000
#define EPAD (EE + NN)          // edges + self loops
#define GG   1024
#define HEADS 4
#define HID  32
#define F1   128                // HEADS*HID
#define EMB  32
#define NCLS 2
#define SLOPE 0.2f
#define BN_EPS 1e-5f

typedef float v2f __attribute__((ext_vector_type(2)));
typedef float v8f __attribute__((ext_vector_type(8)));

// ---------------- helpers ----------------

__device__ __forceinline__ float lrelu(float v) {
    return v > 0.0f ? v : SLOPE * v;
}

// sign-split float atomic max (correct for mixed signs; init with -inf bits)
__device__ __forceinline__ void atomicMaxF(float* addr, float v) {
    if (v >= 0.0f) atomicMax((int*)addr, __float_as_int(v));
    else           atomicMin((unsigned int*)addr, __float_as_uint(v));
}

// ---------------- generic fill ----------------

__global__ void k_fill_u32(unsigned int* __restrict__ p, long long n, unsigned int pat) {
    long long i = (long long)blockIdx.x * blockDim.x + threadIdx.x;
    long long stride = (long long)gridDim.x * blockDim.x;
    for (; i < n; i += stride) p[i] = pat;
}

// ---------------- layer 1: x@W1 fused with attention logits ----------------

__global__ void k_l1_node(const float* __restrict__ x, const float* __restrict__ W1,
                          const float* __restrict__ as1, const float* __restrict__ ad1,
                          float* __restrict__ h1, float* __restrict__ als,
                          float* __restrict__ ald) {
    int n = blockIdx.x * blockDim.x + threadIdx.x;
    if (n >= NN) return;
    float x0 = x[n * 2 + 0], x1 = x[n * 2 + 1];
    float sa[HEADS] = {0.f, 0.f, 0.f, 0.f};
    float sd[HEADS] = {0.f, 0.f, 0.f, 0.f};
    long long base = (long long)n * F1;
    for (int j = 0; j < F1; ++j) {
        float hv = x0 * W1[j] + x1 * W1[F1 + j];
        h1[base + j] = hv;
        int hh = j >> 5, dd = j & 31;
        sa[hh] += hv * as1[hh * HID + dd];
        sd[hh] += hv * ad1[hh * HID + dd];
    }
    for (int hh = 0; hh < HEADS; ++hh) {
        als[n * HEADS + hh] = sa[hh];
        ald[n * HEADS + hh] = sd[hh];
    }
}

// ---------------- edge passes (templated over head count / feature width) ----------------

template <int H>
__global__ void k_edge_max(const int* __restrict__ src, const int* __restrict__ dst,
                           const float* __restrict__ als, const float* __restrict__ ald,
                           float* __restrict__ m) {
    long long t = (long long)blockIdx.x * blockDim.x + threadIdx.x;
    long long total = (long long)EPAD * H;
    if (t >= total) return;
    long long e = t / H; int hh = (int)(t - e * H);
    int s = (e < EE) ? src[e] : (int)(e - EE);
    int d = (e < EE) ? dst[e] : (int)(e - EE);
    float v = lrelu(als[(long long)s * H + hh] + ald[(long long)d * H + hh]);
    atomicMaxF(&m[(long long)d * H + hh], v);
}

template <int H>
__global__ void k_edge_sum(const int* __restrict__ src, const int* __restrict__ dst,
                           const float* __restrict__ als, const float* __restrict__ ald,
                           const float* __restrict__ m, float* __restrict__ z) {
    long long t = (long long)blockIdx.x * blockDim.x + threadIdx.x;
    long long total = (long long)EPAD * H;
    if (t >= total) return;
    long long e = t / H; int hh = (int)(t - e * H);
    int s = (e < EE) ? src[e] : (int)(e - EE);
    int d = (e < EE) ? dst[e] : (int)(e - EE);
    float v = lrelu(als[(long long)s * H + hh] + ald[(long long)d * H + hh]);
    float w = expf(v - m[(long long)d * H + hh]);
    atomicAdd(&z[(long long)d * H + hh], w);
}

template <int H, int D>
__global__ void k_edge_scatter(const int* __restrict__ src, const int* __restrict__ dst,
                               const float* __restrict__ als, const float* __restrict__ ald,
                               const float* __restrict__ m, const float* __restrict__ z,
                               const float* __restrict__ hsrc, float* __restrict__ agg) {
    const int F = H * D;
    long long t = (long long)blockIdx.x * blockDim.x + threadIdx.x;
    long long total = (long long)EPAD * F;
    if (t >= total) return;
    long long e = t / F; int f = (int)(t - e * F);
    int hh = f / D;
    int s = (e < EE) ? src[e] : (int)(e - EE);
    int d = (e < EE) ? dst[e] : (int)(e - EE);
    float v  = lrelu(als[(long long)s * H + hh] + ald[(long long)d * H + hh]);
    float mm = m[(long long)d * H + hh];
    float zz = z[(long long)d * H + hh];
    float alpha = expf(v - mm) / (zz + 1e-16f);
    atomicAdd(&agg[(long long)d * F + f], hsrc[(long long)s * F + f] * alpha);
}

// ---------------- batch norm ----------------

__global__ void k_bn_stats(const float* __restrict__ v, const float* __restrict__ bias,
                           float* __restrict__ mu, float* __restrict__ var, int C) {
    int c = blockIdx.x;            // one block per channel
    float b = bias[c];
    float s = 0.f, s2 = 0.f;
    for (int n = threadIdx.x; n < NN; n += blockDim.x) {
        float val = v[(long long)n * C + c] + b;
        s += val; s2 += val * val;
    }
    __shared__ float sh[256], sh2[256];
    sh[threadIdx.x] = s; sh2[threadIdx.x] = s2;
    __syncthreads();
    for (int o = 128; o > 0; o >>= 1) {
        if ((int)threadIdx.x < o) {
            sh[threadIdx.x]  += sh[threadIdx.x + o];
            sh2[threadIdx.x] += sh2[threadIdx.x + o];
        }
        __syncthreads();
    }
    if (threadIdx.x == 0) {
        float mval = sh[0] / (float)NN;
        mu[c]  = mval;
        var[c] = sh2[0] / (float)NN - mval * mval;
    }
}

__global__ void k_bn_elu(float* __restrict__ v, const float* __restrict__ bias,
                         const float* __restrict__ mu, const float* __restrict__ var,
                         const float* __restrict__ gamma, const float* __restrict__ beta,
                         int C) {
    long long t = (long long)blockIdx.x * blockDim.x + threadIdx.x;
    long long total = (long long)NN * C;
    if (t >= total) return;
    int c = (int)(t % C);
    float val = v[t] + bias[c];
    val = (val - mu[c]) * rsqrtf(var[c] + BN_EPS) * gamma[c] + beta[c];
    v[t] = val > 0.0f ? val : (expf(val) - 1.0f);
}

// ---------------- layer 2 GEMM: [N,128] @ [128,32] via V_WMMA_F32_16X16X4_F32 ----------------
// One wave computes a 16x32 tile: K-loop of 32 steps, 2 column tiles -> 64 WMMA per wave.
// A frag (16x4 f32): lanes 0-15 -> K=k,k+1 ; lanes 16-31 -> K=k+2,k+3 (M = lane&15).
// B frag (4x16 f32): lanes 0-15 -> K=k,k+1 ; lanes 16-31 -> K=k+2,k+3 (N = lane&15).
// C/D (16x16 f32):   VGPR r = row (r + 8*(lane>>4)), col = lane&15.

__global__ void k_gemm2_wmma(const float* __restrict__ X, const float* __restrict__ W,
                             float* __restrict__ Y) {
    const int tilesM = NN / 16;                       // 6250, exact
    int wave = blockIdx.x * (blockDim.x >> 5) + (threadIdx.x >> 5);
    if (wave >= tilesM) return;                       // whole-wave uniform branch
    int lane = threadIdx.x & 31;
    int mbase = wave * 16;
    int rowA  = mbase + (lane & 15);
    int koff  = (lane >> 4) * 2;
    int colB  = lane & 15;

    const float* xrow = X + (long long)rowA * F1;
    v8f c0 = {}; v8f c1 = {};

#pragma unroll 4
    for (int k = 0; k < F1; k += 4) {
        v2f a;
        a.x = xrow[k + koff];
        a.y = xrow[k + koff + 1];
        v2f b0, b1;
        b0.x = W[(k + koff) * EMB + colB];
        b0.y = W[(k + koff + 1) * EMB + colB];
        b1.x = W[(k + koff) * EMB + 16 + colB];
        b1.y = W[(k + koff + 1) * EMB + 16 + colB];
        c0 = __builtin_amdgcn_wmma_f32_16x16x4_f32(false, a, false, b0, (short)0, c0, false, false);
        c1 = __builtin_amdgcn_wmma_f32_16x16x4_f32(false, a, false, b1, (short)0, c1, false, false);
    }

    int rbase = mbase + (lane >> 4) * 8;
    int col = lane & 15;
#pragma unroll
    for (int r = 0; r < 8; ++r) {
        Y[(long long)(rbase + r) * EMB + col]      = c0[r];
        Y[(long long)(rbase + r) * EMB + 16 + col] = c1[r];
    }
}

// ---------------- layer 2 attention logits ----------------

__global__ void k_l2_att(const float* __restrict__ h2, const float* __restrict__ as2,
                         const float* __restrict__ ad2, float* __restrict__ als,
                         float* __restrict__ ald) {
    int n = blockIdx.x * blockDim.x + threadIdx.x;
    if (n >= NN) return;
    float sa = 0.f, sd = 0.f;
    long long base = (long long)n * EMB;
    for (int j = 0; j < EMB; ++j) {
        float v = h2[base + j];
        sa += v * as2[j];
        sd += v * ad2[j];
    }
    als[n] = sa; ald[n] = sd;
}

// ---------------- pooling + classifier ----------------

__global__ void k_pool(const float* __restrict__ hf, const int* __restrict__ batch,
                       float* __restrict__ sums, float* __restrict__ cnt) {
    long long t = (long long)blockIdx.x * blockDim.x + threadIdx.x;
    long long total = (long long)NN * EMB;
    if (t >= total) return;
    int n = (int)(t >> 5);
    int c = (int)(t & 31);
    int g = batch[n];
    atomicAdd(&sums[g * EMB + c], hf[t]);
    if (c == 0) atomicAdd(&cnt[g], 1.0f);
}

__global__ void k_cls(const float* __restrict__ sums, const float* __restrict__ cnt,
                      const float* __restrict__ Wc, const float* __restrict__ bc,
                      float* __restrict__ out) {
    int t = blockIdx.x * blockDim.x + threadIdx.x;
    if (t >= GG * NCLS) return;
    int g = t / NCLS, c = t % NCLS;
    float inv = 1.0f / fmaxf(cnt[g], 1.0f);
    float acc = bc[c];
    for (int d = 0; d < EMB; ++d) acc += (sums[g * EMB + d] * inv) * Wc[d * NCLS + c];
    out[t] = acc;
}

// ---------------- launch ----------------

static inline unsigned int nblk(long long total, int bs) {
    return (unsigned int)((total + bs - 1) / bs);
}

extern "C" void kernel_launch(void* const* d_in, const int* in_sizes, int n_in,
                              void* d_out, int out_size, void* d_ws, size_t ws_size,
                              hipStream_t stream) {
    const float* x    = (const float*)d_in[0];
    const int*   ei   = (const int*)d_in[1];
    const int*   batch= (const int*)d_in[2];
    const float* W1   = (const float*)d_in[3];
    const float* as1  = (const float*)d_in[4];
    const float* ad1  = (const float*)d_in[5];
    const float* b1   = (const float*)d_in[6];
    const float* g1   = (const float*)d_in[7];
    const float* be1  = (const float*)d_in[8];
    const float* W2   = (const float*)d_in[9];
    const float* as2  = (const float*)d_in[10];
    const float* ad2  = (const float*)d_in[11];
    const float* b2   = (const float*)d_in[12];
    const float* g2   = (const float*)d_in[13];
    const float* be2  = (const float*)d_in[14];
    const float* Wc   = (const float*)d_in[15];
    const float* bc   = (const float*)d_in[16];
    const int* srcp = ei;
    const int* dstp = ei + EE;

    float* ws = (float*)d_ws;
    // workspace layout (floats)
    float* bufA  = ws;                              // N*128 : h1, later h2 (N*32) + agg2 (N*32)
    float* bufB  = bufA + (long long)NN * F1;       // N*128 : agg1 -> BN/ELU in place
    float* als1  = bufB + (long long)NN * F1;       // N*4
    float* ald1  = als1 + (long long)NN * HEADS;    // N*4
    float* m1    = ald1 + (long long)NN * HEADS;    // N*4
    float* z1    = m1   + (long long)NN * HEADS;    // N*4
    float* als2  = z1   + (long long)NN * HEADS;    // N
    float* ald2  = als2 + NN;                       // N
    float* m2    = ald2 + NN;                       // N
    float* z2    = m2   + NN;                       // N
    float* mu1   = z2   + NN;                       // 128
    float* var1  = mu1  + F1;                       // 128
    float* mu2   = var1 + F1;                       // 32
    float* var2  = mu2  + EMB;                      // 32
    float* psum  = var2 + EMB;                      // G*32
    float* pcnt  = psum + (long long)GG * EMB;      // G

    float* h1   = bufA;
    float* agg1 = bufB;
    float* h2   = bufA;                             // reuses h1 space after layer-1 done
    float* agg2 = bufA + (long long)NN * EMB;

    const int BS = 256;
    const unsigned int NEG_INF = 0xFF800000u;

    // ---- layer 1 ----
    k_fill_u32<<<nblk((long long)NN * HEADS, BS), BS, 0, stream>>>((unsigned int*)m1, (long long)NN * HEADS, NEG_INF);
    k_fill_u32<<<nblk((long long)NN * HEADS, BS), BS, 0, stream>>>((unsigned int*)z1, (long long)NN * HEADS, 0u);
    k_fill_u32<<<nblk((long long)NN * F1, BS), BS, 0, stream>>>((unsigned int*)agg1, (long long)NN * F1, 0u);

    k_l1_node<<<nblk(NN, BS), BS, 0, stream>>>(x, W1, as1, ad1, h1, als1, ald1);
    k_edge_max<HEADS><<<nblk((long long)EPAD * HEADS, BS), BS, 0, stream>>>(srcp, dstp, als1, ald1, m1);
    k_edge_sum<HEADS><<<nblk((long long)EPAD * HEADS, BS), BS, 0, stream>>>(srcp, dstp, als1, ald1, m1, z1);
    k_edge_scatter<HEADS, HID><<<nblk((long long)EPAD * F1, BS), BS, 0, stream>>>(srcp, dstp, als1, ald1, m1, z1, h1, agg1);

    k_bn_stats<<<F1, BS, 0, stream>>>(agg1, b1, mu1, var1, F1);
    k_bn_elu<<<nblk((long long)NN * F1, BS), BS, 0, stream>>>(agg1, b1, mu1, var1, g1, be1, F1);

    // ---- layer 2 ----
    k_gemm2_wmma<<<nblk(NN / 16, 8), 256, 0, stream>>>(agg1, W2, h2);
    k_l2_att<<<nblk(NN, BS), BS, 0, stream>>>(h2, as2, ad2, als2, ald2);

    k_fill_u32<<<nblk(NN, BS), BS, 0, stream>>>((unsigned int*)m2, NN, NEG_INF);
    k_fill_u32<<<nblk(NN, BS), BS, 0, stream>>>((unsigned int*)z2, NN, 0u);
    k_fill_u32<<<nblk((long long)NN * EMB, BS), BS, 0, stream>>>((unsigned int*)agg2, (long long)NN * EMB, 0u);

    k_edge_max<1><<<nblk((long long)EPAD, BS), BS, 0, stream>>>(srcp, dstp, als2, ald2, m2);
    k_edge_sum<1><<<nblk((long long)EPAD, BS), BS, 0, stream>>>(srcp, dstp, als2, ald2, m2, z2);
    k_edge_scatter<1, EMB><<<nblk((long long)EPAD * EMB, BS), BS, 0, stream>>>(srcp, dstp, als2, ald2, m2, z2, h2, agg2);

    k_bn_stats<<<EMB, BS, 0, stream>>>(agg2, b2, mu2, var2, EMB);
    k_bn_elu<<<nblk((long long)NN * EMB, BS), BS, 0, stream>>>(agg2, b2, mu2, var2, g2, be2, EMB);

    // ---- pool + classifier ----
    k_fill_u32<<<nblk((long long)GG * EMB + GG, BS), BS, 0, stream>>>((unsigned int*)psum, (long long)GG * EMB + GG, 0u);
    k_pool<<<nblk((long long)NN * EMB, BS), BS, 0, stream>>>(agg2, batch, psum, pcnt);
    k_cls<<<nblk(GG * NCLS, BS), BS, 0, stream>>>(psum, pcnt, Wc, bc, (float*)d_out);
}